// TsConfidence_47553877901972
// MI455X (gfx1250) — compile-verified
//
#include <hip/hip_runtime.h>
#include <math.h>

// ---------------------------------------------------------------------------
// TsConfidence forward, gfx1250 (CDNA5, wave32).
// Dead-code-eliminated: x_vec / e_vec / coordinate updates never reach the
// output, so only the x_sca / e_sca scalar chain is computed.
// GEMM-shaped ops use V_WMMA_F32_16X16X4_F32 (fp32 in/out). Fast path:
// software-pipelined loads (next K-step's A + NT B-frags loaded before the
// current step's NT back-to-back WMMAs), strength-reduced B addressing.
// Guarded path (odd K/N): clamped addresses + value select, no exec churn.
// ---------------------------------------------------------------------------

typedef float v2f __attribute__((ext_vector_type(2)));
typedef float v8f __attribute__((ext_vector_type(8)));

#define Bg   16
#define Nn   128
#define DSd  128
#define DESd 64
#define Hh   8
#define DHd  16
#define ROWS  (Bg * Nn)        // 2048 node rows
#define EROWS (Bg * Nn * Nn)   // 262144 edge rows

enum { ACT_NONE = 0, ACT_SELU = 1, ACT_SIG = 2, ACT_GATE = 3 };

__device__ __forceinline__ float selu_f(float x) {
  const float a = 1.6732632423543772f, s = 1.0507009873554805f;
  return x > 0.f ? s * x : s * a * (__expf(x) - 1.f);
}
__device__ __forceinline__ float sigm_f(float x) { return 1.f / (1.f + __expf(-x)); }

// WMMA f32 16x16x4 lane layout (wave32):
//   A frag (16x4):  row = lane&15, k = k0 + 2*(lane>>4) + j   (j = vgpr 0,1)
//   B frag (4x16):  col = lane&15, k = k0 + 2*(lane>>4) + v   (v = vgpr 0,1)
//   C (16x16):      row = v + 8*(lane>>4), col = lane&15      (v = vgpr 0..7)

// Fast path: K%4==0, all columns valid. 1-deep software pipeline, peeled tail.
template <int NT>
__device__ __forceinline__ void gemm_accum_fast(v8f* acc, const float* __restrict__ Ar,
                                                const float* __restrict__ Bm, int ldb,
                                                int K, int ncol, int kh) {
  const float* ap = Ar + kh;
  const float* b0 = Bm + (size_t)kh * ldb + ncol;
  const float* b1 = b0 + ldb;
  const size_t bstep = (size_t)4 * ldb;
  __builtin_prefetch(ap + 16, 0, 0);

  v2f a = *(const v2f*)ap;                       // one 64-bit load
  v2f bf[NT];
#pragma unroll
  for (int t = 0; t < NT; ++t) { bf[t][0] = b0[16 * t]; bf[t][1] = b1[16 * t]; }

#pragma unroll 2
  for (int k0 = 0; k0 < K - 4; k0 += 4) {
    ap += 4; b0 += bstep; b1 += bstep;
    v2f a_n = *(const v2f*)ap;                   // prefetch next step's frags
    v2f bf_n[NT];
#pragma unroll
    for (int t = 0; t < NT; ++t) { bf_n[t][0] = b0[16 * t]; bf_n[t][1] = b1[16 * t]; }
#pragma unroll
    for (int t = 0; t < NT; ++t)                 // NT back-to-back WMMAs
      acc[t] = __builtin_amdgcn_wmma_f32_16x16x4_f32(false, a, false, bf[t],
                                                     (short)0, acc[t], false, false);
    a = a_n;
#pragma unroll
    for (int t = 0; t < NT; ++t) bf[t] = bf_n[t];
  }
#pragma unroll
  for (int t = 0; t < NT; ++t)                   // peeled last step
    acc[t] = __builtin_amdgcn_wmma_f32_16x16x4_f32(false, a, false, bf[t],
                                                   (short)0, acc[t], false, false);
}

// Guarded path: clamp indices, select values (no exec-mask branching).
__device__ __forceinline__ void gemm_accum_guard(v8f* acc, const float* __restrict__ Ar,
                                                 const float* __restrict__ Bm, int ldb,
                                                 int K, int ncol, int kh, int N) {
  for (int k0 = 0; k0 < K; k0 += 4) {
    int ka = k0 + kh;
    int c0 = ka < K ? ka : 0;
    int c1 = ka + 1 < K ? ka + 1 : 0;
    float a0 = Ar[c0], a1 = Ar[c1];
    v2f a;
    a[0] = ka < K ? a0 : 0.f;
    a[1] = ka + 1 < K ? a1 : 0.f;
    int nc = ncol < N ? ncol : 0;
    float b0 = Bm[(size_t)c0 * ldb + nc];
    float b1 = Bm[(size_t)c1 * ldb + nc];
    v2f b;
    b[0] = (ka < K && ncol < N) ? b0 : 0.f;
    b[1] = (ka + 1 < K && ncol < N) ? b1 : 0.f;
    acc[0] = __builtin_amdgcn_wmma_f32_16x16x4_f32(false, a, false, b,
                                                   (short)0, acc[0], false, false);
  }
}

// C = act( A@B [+ A2@B2] + bias ), row-major. A:[M,K] lda, B:[K,N] ldb,
// A2:[M,K2] (lda2==K2), B2:[K2,N] ldb. ACT_GATE: g=sigmoid(acc+bias),
// C = g*R + (1-g)*P. One wave computes a 16 x (16*NT) band.
template <int NT, bool GUARD>
__global__ void gemm_wmma_kernel(const float* __restrict__ A, int lda,
                                 const float* __restrict__ Bm, int ldb, int K,
                                 const float* __restrict__ A2,
                                 const float* __restrict__ B2, int K2,
                                 const float* __restrict__ bias,
                                 const float* __restrict__ Rg,
                                 const float* __restrict__ Pg,
                                 float* __restrict__ C, int ldc,
                                 int M, int N, int act, int waves, int wavesN) {
  int gw = (int)((blockIdx.x * blockDim.x + threadIdx.x) >> 5);
  if (gw >= waves) return;
  int lane = threadIdx.x & 31;
  int tm = gw / wavesN;
  int tg = gw - tm * wavesN;
  int row16 = tm * 16;
  int ncol = tg * (16 * NT) + (lane & 15);
  int kh = (lane >> 4) << 1;  // 0 or 2

  v8f acc[NT];
#pragma unroll
  for (int t = 0; t < NT; ++t) acc[t] = {0.f, 0.f, 0.f, 0.f, 0.f, 0.f, 0.f, 0.f};

  const float* Ar = A + (size_t)(row16 + (lane & 15)) * lda;
  if (GUARD) {
    gemm_accum_guard(acc, Ar, Bm, ldb, K, ncol, kh, N);
  } else {
    gemm_accum_fast<NT>(acc, Ar, Bm, ldb, K, ncol, kh);
  }
  if (A2) {
    const float* A2r = A2 + (size_t)(row16 + (lane & 15)) * K2;
    if (GUARD) {
      gemm_accum_guard(acc, A2r, B2, ldb, K2, ncol, kh, N);
    } else {
      gemm_accum_fast<NT>(acc, A2r, B2, ldb, K2, ncol, kh);
    }
  }

#pragma unroll
  for (int t = 0; t < NT; ++t) {
    int n = ncol + 16 * t;
    if (GUARD && n >= N) continue;
#pragma unroll
    for (int v = 0; v < 8; ++v) {
      int rr = row16 + v + ((lane >> 4) << 3);
      float val = acc[t][v];
      if (bias) val += bias[n];
      if (act == ACT_SELU) {
        val = selu_f(val);
      } else if (act == ACT_SIG) {
        val = sigm_f(val);
      } else if (act == ACT_GATE) {
        float g = sigm_f(val);
        size_t idx = (size_t)rr * ldc + n;
        val = g * Rg[idx] + (1.f - g) * Pg[idx];
      }
      C[(size_t)rr * ldc + n] = val;
    }
  }
}

// ---------------------------------------------------------------------------
// LayerNorm over residual: O = LN( X + (G? G*Y : Y) ) * S + Bb.  cols == 128.
// One wave per row, 4 elems/lane, wave32 shfl_xor reductions.
// ---------------------------------------------------------------------------
__global__ void ln_kernel(const float* __restrict__ X, const float* __restrict__ Y,
                          const float* __restrict__ G, const float* __restrict__ S,
                          const float* __restrict__ Bb, float* __restrict__ O,
                          int rows) {
  int wave = threadIdx.x >> 5, lane = threadIdx.x & 31;
  int row = blockIdx.x * (blockDim.x >> 5) + wave;
  if (row >= rows) return;
  const float4 xv = ((const float4*)(X + (size_t)row * 128))[lane];
  const float4 yv = ((const float4*)(Y + (size_t)row * 128))[lane];
  float r0, r1, r2, r3;
  if (G) {
    const float4 gv = ((const float4*)(G + (size_t)row * 128))[lane];
    r0 = xv.x + gv.x * yv.x; r1 = xv.y + gv.y * yv.y;
    r2 = xv.z + gv.z * yv.z; r3 = xv.w + gv.w * yv.w;
  } else {
    r0 = xv.x + yv.x; r1 = xv.y + yv.y; r2 = xv.z + yv.z; r3 = xv.w + yv.w;
  }
  float sum = r0 + r1 + r2 + r3;
  for (int o = 16; o >= 1; o >>= 1) sum += __shfl_xor(sum, o, 32);
  float mean = sum * (1.f / 128.f);
  float d0 = r0 - mean, d1 = r1 - mean, d2 = r2 - mean, d3 = r3 - mean;
  float vs = d0 * d0 + d1 * d1 + d2 * d2 + d3 * d3;
  for (int o = 16; o >= 1; o >>= 1) vs += __shfl_xor(vs, o, 32);
  float inv = rsqrtf(vs * (1.f / 128.f) + 1e-5f);
  const float4 sv = ((const float4*)S)[lane];
  const float4 bv = ((const float4*)Bb)[lane];
  float4 ov;
  ov.x = d0 * inv * sv.x + bv.x;
  ov.y = d1 * inv * sv.y + bv.y;
  ov.z = d2 * inv * sv.z + bv.z;
  ov.w = d3 * inv * sv.w + bv.w;
  ((float4*)(O + (size_t)row * 128))[lane] = ov;
}

// ---------------------------------------------------------------------------
// Masked attention, one block per (b,h). K/V head-slices staged in LDS.
// Two-pass streaming softmax. DH = 16.
// ---------------------------------------------------------------------------
__global__ void attn_kernel(const float* __restrict__ Q, const float* __restrict__ Km,
                            const float* __restrict__ Vm, const float* __restrict__ biasb,
                            const int* __restrict__ masks, float* __restrict__ Out) {
  __shared__ float Ks[Nn][DHd + 1];
  __shared__ float Vs[Nn][DHd + 1];
  __shared__ float neg[Nn];
  int b = blockIdx.x >> 3;
  int h = blockIdx.x & 7;
  int i = threadIdx.x;  // 0..127
  {
    const float* kp = Km + ((size_t)(b * Nn + i) * DSd) + h * DHd;
    const float* vp = Vm + ((size_t)(b * Nn + i) * DSd) + h * DHd;
#pragma unroll
    for (int d = 0; d < DHd; ++d) { Ks[i][d] = kp[d]; Vs[i][d] = vp[d]; }
    neg[i] = (masks[b * Nn + i] > 0) ? 0.f : -1e9f;
  }
  __syncthreads();
  float q[DHd];
  const float* qp = Q + ((size_t)(b * Nn + i) * DSd) + h * DHd;
#pragma unroll
  for (int d = 0; d < DHd; ++d) q[d] = qp[d];
  const float* bp = biasb + ((size_t)(b * Nn + i) * Nn) * Hh + h;

  float mx = -1e30f;
  for (int j = 0; j < Nn; ++j) {
    float s = 0.f;
#pragma unroll
    for (int d = 0; d < DHd; ++d) s += q[d] * Ks[j][d];
    s = s * 0.25f + bp[(size_t)j * Hh] + neg[j];
    mx = fmaxf(mx, s);
  }
  float l = 0.f;
  float o[DHd];
#pragma unroll
  for (int d = 0; d < DHd; ++d) o[d] = 0.f;
  for (int j = 0; j < Nn; ++j) {
    float s = 0.f;
#pragma unroll
    for (int d = 0; d < DHd; ++d) s += q[d] * Ks[j][d];
    s = s * 0.25f + bp[(size_t)j * Hh] + neg[j];
    float p = __expf(s - mx);
    l += p;
#pragma unroll
    for (int d = 0; d < DHd; ++d) o[d] += p * Vs[j][d];
  }
  float inv = 1.f / l;
  float* op = Out + ((size_t)(b * Nn + i) * DSd) + h * DHd;
#pragma unroll
  for (int d = 0; d < DHd; ++d) op[d] = o[d] * inv;
}

// ---------------------------------------------------------------------------
// e[b,i,j,:] += pair[b,i,:] + pair[b,j,:]   (float4 over DES=64)
// ---------------------------------------------------------------------------
__global__ void edge_update_kernel(float* __restrict__ E, const float* __restrict__ Pair) {
  size_t idx = (size_t)blockIdx.x * blockDim.x + threadIdx.x;  // float4 index
  const size_t total = (size_t)EROWS * (DESd / 4);
  if (idx >= total) return;
  int c4 = (int)(idx & 15);
  size_t pidx = idx >> 4;          // (b,i,j)
  int j = (int)(pidx & (Nn - 1));
  size_t bi = pidx >> 7;           // b*Nn + i
  int b = (int)(bi >> 7);
  float4 a = ((const float4*)(Pair + bi * DESd))[c4];
  float4 c = ((const float4*)(Pair + ((size_t)b * Nn + j) * DESd))[c4];
  float4* ep = (float4*)E + idx;
  float4 ev = *ep;
  ev.x += a.x + c.x; ev.y += a.y + c.y; ev.z += a.z + c.z; ev.w += a.w + c.w;
  *ep = ev;
}

// ---------------------------------------------------------------------------
// Readout: out[b] = sum_n softmax_n(att[b,n] - (1-mask)*1e6) * emb[b,n]
// ---------------------------------------------------------------------------
__global__ void pool_kernel(const float* __restrict__ att_n, const float* __restrict__ emb_n,
                            const int* __restrict__ masks, float* __restrict__ out) {
  __shared__ float red[Nn];
  int b = blockIdx.x, n = threadIdx.x;
  float e = att_n[b * Nn + n] - ((masks[b * Nn + n] > 0) ? 0.f : 1e6f);
  red[n] = e;
  __syncthreads();
  for (int s = 64; s > 0; s >>= 1) { if (n < s) red[n] = fmaxf(red[n], red[n + s]); __syncthreads(); }
  float mx = red[0];
  __syncthreads();
  float p = __expf(e - mx);
  red[n] = p;
  __syncthreads();
  for (int s = 64; s > 0; s >>= 1) { if (n < s) red[n] += red[n + s]; __syncthreads(); }
  float sp = red[0];
  __syncthreads();
  red[n] = p * emb_n[b * Nn + n];
  __syncthreads();
  for (int s = 64; s > 0; s >>= 1) { if (n < s) red[n] += red[n + s]; __syncthreads(); }
  if (n == 0) out[b] = red[0] / sp;
}

// ---------------------------------------------------------------------------
// Host side
// ---------------------------------------------------------------------------
struct LayerP {
  const float *W1, *W2, *Web, *Wei, *Wg, *Wk, *Wq, *Wv;
  const float *b1, *b2, *be, *bg, *ln1b, *ln1s, *ln2b, *ln2s;
};

// params pytree leaf indices (top-level dict in insertion order; `params`
// flattened with JAX sorted-key order):
//   inputs: 0 ratoms, 1 patoms, 2 radjs, 3 padjs, 4-6 coords (unused), 7 masks
//   params: 8.. att(10), 18 egate(Ws,Wv,bs,bv), 22 emb(10),
//           32 penc{eeW,eeb,eevW,evW,exW,exb,layers@38+19l},
//           114 renc{...,layers@120+19l}, 196 ts_layers(4x19),
//           272 tsmlp(6), 278 xgate(Ws,Wv,bs,bv)
// per-layer sorted keys: W1,W2,Wcoor,Web,Wei,Wg,Wgv,Wk,Wq,Wv,Wvm,
//                        b1,b2,be,bg,ln1b,ln1s,ln2b,ln2s
static inline const float* FP(void* const* d, int i) { return (const float*)d[i]; }

static LayerP layer_at(void* const* d, int base) {
  LayerP L;
  L.W1 = FP(d, base + 0);  L.W2 = FP(d, base + 1);
  L.Web = FP(d, base + 3); L.Wei = FP(d, base + 4); L.Wg = FP(d, base + 5);
  L.Wk = FP(d, base + 7);  L.Wq = FP(d, base + 8);  L.Wv = FP(d, base + 9);
  L.b1 = FP(d, base + 11); L.b2 = FP(d, base + 12);
  L.be = FP(d, base + 13); L.bg = FP(d, base + 14);
  L.ln1b = FP(d, base + 15); L.ln1s = FP(d, base + 16);
  L.ln2b = FP(d, base + 17); L.ln2s = FP(d, base + 18);
  return L;
}

extern "C" void kernel_launch(void* const* d_in, const int* in_sizes, int n_in,
                              void* d_out, int out_size, void* d_ws, size_t ws_size,
                              hipStream_t stream) {
  (void)in_sizes; (void)n_in; (void)out_size; (void)ws_size;
  const float* ratoms = FP(d_in, 0);
  const float* patoms = FP(d_in, 1);
  const float* radjs  = FP(d_in, 2);
  const float* padjs  = FP(d_in, 3);
  const int*   masks  = (const int*)d_in[7];

  // ---- workspace partition -------------------------------------------------
  size_t off = 0;
  auto wsf = [&](size_t elems) {
    float* p = (float*)((char*)d_ws + off);
    off += ((elems * 4) + 255) & ~(size_t)255;
    return p;
  };
  float* r_es  = wsf((size_t)EROWS * DESd);   // 64 MB
  float* p_es  = wsf((size_t)EROWS * DESd);   // 64 MB
  float* ecomb = wsf((size_t)EROWS * DESd);   // 64 MB
  float* biasb = wsf((size_t)EROWS * Hh);     // 8 MB
  float* r_xs  = wsf((size_t)ROWS * DSd);
  float* p_xs  = wsf((size_t)ROWS * DSd);
  float* xcomb = wsf((size_t)ROWS * DSd);
  float* Qb    = wsf((size_t)ROWS * DSd);
  float* Kb    = wsf((size_t)ROWS * DSd);
  float* Vb    = wsf((size_t)ROWS * DSd);
  float* Gb    = wsf((size_t)ROWS * DSd);
  float* attno = wsf((size_t)ROWS * DSd);
  float* x1b   = wsf((size_t)ROWS * DSd);
  float* t1b   = wsf((size_t)ROWS * 2 * DSd);
  float* x2p   = wsf((size_t)ROWS * DSd);
  float* pairb = wsf((size_t)ROWS * DESd);
  float* hb    = wsf((size_t)ROWS * DSd);
  float* mA    = wsf((size_t)ROWS * DSd);
  float* mB    = wsf((size_t)ROWS * DSd);
  float* att_n = wsf((size_t)ROWS);
  float* emb_n = wsf((size_t)ROWS);

  // ---- launch helpers ------------------------------------------------------
  auto gemm = [&](const float* A, int lda, const float* Bm, int ldb, int K,
                  const float* A2, const float* B2, int K2,
                  const float* bias, const float* R, const float* P,
                  float* C, int ldc, int M, int N, int act) {
    int tilesM = M / 16;
    bool fast = ((K & 3) == 0) && ((N & 63) == 0) && (A2 == nullptr || (K2 & 3) == 0);
    if (fast) {
      int wavesN = N / 64;                 // NT = 4 -> 16x64 band per wave
      int waves = tilesM * wavesN;
      int blocks = (waves + 3) / 4;
      gemm_wmma_kernel<4, false><<<blocks, 128, 0, stream>>>(
          A, lda, Bm, ldb, K, A2, B2, K2, bias, R, P, C, ldc, M, N, act, waves, wavesN);
    } else {
      int wavesN = (N + 15) / 16;
      int waves = tilesM * wavesN;
      int blocks = (waves + 3) / 4;
      gemm_wmma_kernel<1, true><<<blocks, 128, 0, stream>>>(
          A, lda, Bm, ldb, K, A2, B2, K2, bias, R, P, C, ldc, M, N, act, waves, wavesN);
    }
  };
  auto lnorm = [&](const float* X, const float* Y, const float* G,
                   const float* S, const float* Bb, float* O) {
    ln_kernel<<<ROWS / 8, 256, 0, stream>>>(X, Y, G, S, Bb, O, ROWS);
  };
  auto run_layer = [&](const LayerP& L, float* x, float* e) {
    gemm(x, DSd, L.Wq, DSd, DSd, nullptr, nullptr, 0, nullptr, nullptr, nullptr,
         Qb, DSd, ROWS, DSd, ACT_NONE);
    gemm(x, DSd, L.Wk, DSd, DSd, nullptr, nullptr, 0, nullptr, nullptr, nullptr,
         Kb, DSd, ROWS, DSd, ACT_NONE);
    gemm(x, DSd, L.Wv, DSd, DSd, nullptr, nullptr, 0, nullptr, nullptr, nullptr,
         Vb, DSd, ROWS, DSd, ACT_NONE);
    gemm(x, DSd, L.Wg, DSd, DSd, nullptr, nullptr, 0, L.bg, nullptr, nullptr,
         Gb, DSd, ROWS, DSd, ACT_SIG);
    // edge bias: [B*N*N,64] @ [64,8]  (N=8 -> guarded path)
    gemm(e, DESd, L.Web, Hh, DESd, nullptr, nullptr, 0, nullptr, nullptr, nullptr,
         biasb, Hh, EROWS, Hh, ACT_NONE);
    attn_kernel<<<Bg * Hh, Nn, 0, stream>>>(Qb, Kb, Vb, biasb, masks, attno);
    lnorm(x, attno, Gb, L.ln1s, L.ln1b, x1b);                 // x1 = LN(x + g*out)
    gemm(x1b, DSd, L.W1, 2 * DSd, DSd, nullptr, nullptr, 0, L.b1, nullptr, nullptr,
         t1b, 2 * DSd, ROWS, 2 * DSd, ACT_SELU);
    gemm(t1b, 2 * DSd, L.W2, DSd, 2 * DSd, nullptr, nullptr, 0, L.b2, nullptr, nullptr,
         x2p, DSd, ROWS, DSd, ACT_NONE);
    lnorm(x1b, x2p, nullptr, L.ln2s, L.ln2b, x);              // x = LN(x1 + ffn)
    gemm(x, DSd, L.Wei, DESd, DSd, nullptr, nullptr, 0, L.be, nullptr, nullptr,
         pairb, DESd, ROWS, DESd, ACT_NONE);
    edge_update_kernel<<<(EROWS * (DESd / 4) + 255) / 256, 256, 0, stream>>>(e, pairb);
  };

  // ---- parameter pointers --------------------------------------------------
  const int PENC = 32, RENC = 114, TS = 196;
  const float* renc_eeW = FP(d_in, RENC + 0); const float* renc_eeb = FP(d_in, RENC + 1);
  const float* renc_exW = FP(d_in, RENC + 4); const float* renc_exb = FP(d_in, RENC + 5);
  const float* penc_eeW = FP(d_in, PENC + 0); const float* penc_eeb = FP(d_in, PENC + 1);
  const float* penc_exW = FP(d_in, PENC + 4); const float* penc_exb = FP(d_in, PENC + 5);
  const float* eg_Ws = FP(d_in, 18); const float* eg_bs = FP(d_in, 20);
  const float* xg_Ws = FP(d_in, 278); const float* xg_bs = FP(d_in, 280);

  // ---- stage 0: embeddings (note r/p encoder swap from reference) ----------
  gemm(patoms, 30, renc_exW, DSd, 30, nullptr, nullptr, 0, renc_exb, nullptr, nullptr,
       p_xs, DSd, ROWS, DSd, ACT_NONE);
  gemm(padjs, 5, renc_eeW, DESd, 5, nullptr, nullptr, 0, renc_eeb, nullptr, nullptr,
       p_es, DESd, EROWS, DESd, ACT_NONE);
  gemm(ratoms, 30, penc_exW, DSd, 30, nullptr, nullptr, 0, penc_exb, nullptr, nullptr,
       r_xs, DSd, ROWS, DSd, ACT_NONE);
  gemm(radjs, 5, penc_eeW, DESd, 5, nullptr, nullptr, 0, penc_eeb, nullptr, nullptr,
       r_es, DESd, EROWS, DESd, ACT_NONE);

  // ---- stage 1: encoder GVT layers ----------------------------------------
  for (int l = 0; l < 4; ++l) { LayerP L = layer_at(d_in, RENC + 6 + 19 * l); run_layer(L, p_xs, p_es); }
  for (int l = 0; l < 4; ++l) { LayerP L = layer_at(d_in, PENC + 6 + 19 * l); run_layer(L, r_xs, r_es); }

  // ---- stage 2: gate residues (concat via dual-A GEMM) ---------------------
  gemm(r_xs, DSd, xg_Ws, DSd, DSd, p_xs, xg_Ws + DSd * DSd, DSd,
       xg_bs, r_xs, p_xs, xcomb, DSd, ROWS, DSd, ACT_GATE);
  gemm(r_es, DESd, eg_Ws, DESd, DESd, p_es, eg_Ws + DESd * DESd, DESd,
       eg_bs, r_es, p_es, ecomb, DESd, EROWS, DESd, ACT_GATE);

  // ---- stage 3: TS GVT layers ---------------------------------------------
  for (int l = 0; l < 4; ++l) { LayerP L = layer_at(d_in, TS + 19 * l); run_layer(L, xcomb, ecomb); }

  // ---- stage 4: tsmlp (3x selu linear, DS->DS) -----------------------------
  gemm(xcomb, DSd, FP(d_in, 272), DSd, DSd, nullptr, nullptr, 0, FP(d_in, 273),
       nullptr, nullptr, mA, DSd, ROWS, DSd, ACT_SELU);
  gemm(mA, DSd, FP(d_in, 274), DSd, DSd, nullptr, nullptr, 0, FP(d_in, 275),
       nullptr, nullptr, mB, DSd, ROWS, DSd, ACT_SELU);
  gemm(mB, DSd, FP(d_in, 276), DSd, DSd, nullptr, nullptr, 0, FP(d_in, 277),
       nullptr, nullptr, hb, DSd, ROWS, DSd, ACT_SELU);

  // ---- stage 5: att MLP on [h, x_sca]  (5 selu layers, last N=1) -----------
  gemm(hb, DSd, FP(d_in, 8), DSd, DSd, xcomb, FP(d_in, 8) + DSd * DSd, DSd,
       FP(d_in, 9), nullptr, nullptr, mA, DSd, ROWS, DSd, ACT_SELU);
  gemm(mA, DSd, FP(d_in, 10), DSd, DSd, nullptr, nullptr, 0, FP(d_in, 11),
       nullptr, nullptr, mB, DSd, ROWS, DSd, ACT_SELU);
  gemm(mB, DSd, FP(d_in, 12), DSd, DSd, nullptr, nullptr, 0, FP(d_in, 13),
       nullptr, nullptr, mA, DSd, ROWS, DSd, ACT_SELU);
  gemm(mA, DSd, FP(d_in, 14), DSd, DSd, nullptr, nullptr, 0, FP(d_in, 15),
       nullptr, nullptr, mB, DSd, ROWS, DSd, ACT_SELU);
  gemm(mB, DSd, FP(d_in, 16), 1, DSd, nullptr, nullptr, 0, FP(d_in, 17),
       nullptr, nullptr, att_n, 1, ROWS, 1, ACT_SELU);

  // ---- stage 6: emb MLP on h  (5 selu layers, last N=1) --------------------
  gemm(hb, DSd, FP(d_in, 22), DSd, DSd, nullptr, nullptr, 0, FP(d_in, 23),
       nullptr, nullptr, mA, DSd, ROWS, DSd, ACT_SELU);
  gemm(mA, DSd, FP(d_in, 24), DSd, DSd, nullptr, nullptr, 0, FP(d_in, 25),
       nullptr, nullptr, mB, DSd, ROWS, DSd, ACT_SELU);
  gemm(mB, DSd, FP(d_in, 26), DSd, DSd, nullptr, nullptr, 0, FP(d_in, 27),
       nullptr, nullptr, mA, DSd, ROWS, DSd, ACT_SELU);
  gemm(mA, DSd, FP(d_in, 28), DSd, DSd, nullptr, nullptr, 0, FP(d_in, 29),
       nullptr, nullptr, mB, DSd, ROWS, DSd, ACT_SELU);
  gemm(mB, DSd, FP(d_in, 30), 1, DSd, nullptr, nullptr, 0, FP(d_in, 31),
       nullptr, nullptr, emb_n, 1, ROWS, 1, ACT_SELU);

  // ---- stage 7: masked softmax pooling ------------------------------------
  pool_kernel<<<Bg, Nn, 0, stream>>>(att_n, emb_n, masks, (float*)d_out);
}